// PointPatchEmbed_52003464020568
// MI455X (gfx1250) — compile-verified
//
#include <hip/hip_runtime.h>

#define B_   8
#define N_   16384
#define NG   512
#define GS   32
#define EMB  384
#define ROWS (B_*NG*GS)      // 131072
#define BN_EPS 1e-5f

typedef float v2f __attribute__((ext_vector_type(2)));
typedef float v8f __attribute__((ext_vector_type(8)));

// ---------------- deterministic float atomic max/min (exact ops) -------------
__device__ inline void atomicMaxFloat(float* addr, float val) {
  if (val >= 0.f) atomicMax((int*)addr, __float_as_int(val));
  else            atomicMin((unsigned int*)addr, __float_as_uint(val));
}
__device__ inline void atomicMinFloat(float* addr, float val) {
  if (val >= 0.f) atomicMin((int*)addr, __float_as_int(val));
  else            atomicMax((unsigned int*)addr, __float_as_uint(val));
}

// ---------------- init max/min buffers --------------------------------------
__global__ void init_kernel(float* __restrict__ gmax, float* __restrict__ gmin) {
  int i = blockIdx.x * 256 + threadIdx.x;
  if (i < B_*NG*EMB) { gmax[i] = -3.402823466e38f; gmin[i] = 3.402823466e38f; }
}

// ---------------- FPS: one block per batch, all state in registers ----------
__global__ __launch_bounds__(1024) void fps_kernel(const float* __restrict__ pts,
                                                   float* __restrict__ cent,
                                                   float* __restrict__ out_cent) {
  const int b = blockIdx.x;
  const int t = threadIdx.x;
  const float* p = pts + (size_t)b * N_ * 3;

  float px[16], py[16], pz[16], dmin[16];
#pragma unroll
  for (int j = 0; j < 16; ++j) {
    int idx = t + j * 1024;
    px[j] = p[idx*3+0]; py[j] = p[idx*3+1]; pz[j] = p[idx*3+2];
    dmin[j] = 1e10f;
  }
  __shared__ float s_cx, s_cy, s_cz;
  __shared__ float s_wv[32];
  __shared__ int   s_wi[32];
  if (t == 0) { s_cx = p[0]; s_cy = p[1]; s_cz = p[2]; }
  __syncthreads();
  float cx = s_cx, cy = s_cy, cz = s_cz;

  for (int it = 0; it < NG; ++it) {
    if (t == 0) {
      int o = (b*NG + it)*3;
      cent[o] = cx; cent[o+1] = cy; cent[o+2] = cz;
      out_cent[o] = cx; out_cent[o+1] = cy; out_cent[o+2] = cz;
    }
    float best = -1.0f; int bi = 0;
#pragma unroll
    for (int j = 0; j < 16; ++j) {
      float dx = px[j]-cx, dy = py[j]-cy, dz = pz[j]-cz;
      float d = dx*dx + dy*dy + dz*dz;
      d = fminf(dmin[j], d);
      dmin[j] = d;
      int idx = t + j * 1024;
      if (d > best || (d == best && idx < bi)) { best = d; bi = idx; }
    }
    // wave32 argmax (tie -> lower index)
#pragma unroll
    for (int s = 16; s > 0; s >>= 1) {
      float ov = __shfl_xor(best, s, 32);
      int   oi = __shfl_xor(bi,   s, 32);
      if (ov > best || (ov == best && oi < bi)) { best = ov; bi = oi; }
    }
    if ((t & 31) == 0) { s_wv[t >> 5] = best; s_wi[t >> 5] = bi; }
    __syncthreads();
    if (t < 32) {
      best = s_wv[t]; bi = s_wi[t];
#pragma unroll
      for (int s = 16; s > 0; s >>= 1) {
        float ov = __shfl_xor(best, s, 32);
        int   oi = __shfl_xor(bi,   s, 32);
        if (ov > best || (ov == best && oi < bi)) { best = ov; bi = oi; }
      }
      if (t == 0) { s_cx = p[bi*3+0]; s_cy = p[bi*3+1]; s_cz = p[bi*3+2]; }
    }
    __syncthreads();
    cx = s_cx; cy = s_cy; cz = s_cz;
  }
}

// ---------------- 32-NN: one block per centroid, distances in registers -----
__global__ __launch_bounds__(256) void knn_kernel(const float* __restrict__ pts,
                                                  const float* __restrict__ cent,
                                                  int* __restrict__ knn) {
  const int bid = blockIdx.x;           // b*NG + m
  const int b = bid / NG;
  const int t = threadIdx.x;
  const float* p = pts + (size_t)b * N_ * 3;
  const float cx = cent[bid*3+0], cy = cent[bid*3+1], cz = cent[bid*3+2];

  float d2[64];
  unsigned long long taken = 0ULL;
#pragma unroll
  for (int j = 0; j < 64; ++j) {
    int idx = t + j * 256;
    float dx = p[idx*3+0]-cx, dy = p[idx*3+1]-cy, dz = p[idx*3+2]-cz;
    d2[j] = dx*dx + dy*dy + dz*dz;
  }
  __shared__ float s_wv[8];
  __shared__ int   s_wi[8];
  __shared__ int   s_win;

  for (int it = 0; it < GS; ++it) {
    float best = 1e30f; int bi = N_;
#pragma unroll
    for (int j = 0; j < 64; ++j) {
      if (!((taken >> j) & 1ULL)) {
        int idx = t + j * 256;
        float d = d2[j];
        if (d < best || (d == best && idx < bi)) { best = d; bi = idx; }
      }
    }
#pragma unroll
    for (int s = 16; s > 0; s >>= 1) {
      float ov = __shfl_xor(best, s, 32);
      int   oi = __shfl_xor(bi,   s, 32);
      if (ov < best || (ov == best && oi < bi)) { best = ov; bi = oi; }
    }
    if ((t & 31) == 0) { s_wv[t >> 5] = best; s_wi[t >> 5] = bi; }
    __syncthreads();
    if (t == 0) {
      float bv = s_wv[0]; int bidx = s_wi[0];
#pragma unroll
      for (int w = 1; w < 8; ++w)
        if (s_wv[w] < bv || (s_wv[w] == bv && s_wi[w] < bidx)) { bv = s_wv[w]; bidx = s_wi[w]; }
      s_win = bidx;
      knn[bid*GS + it] = bidx;
    }
    __syncthreads();
    int win = s_win;
    if ((win & 255) == t) taken |= (1ULL << (win >> 8));
    __syncthreads();
  }
}

// ---------------- layer1: gather + center + 3->64 matmul + bias -------------
__global__ __launch_bounds__(256) void layer1_kernel(const float* __restrict__ pts,
    const float* __restrict__ cent, const int* __restrict__ knn,
    const float* __restrict__ W1, const float* __restrict__ b1,
    float* __restrict__ h1) {
  const int tid = blockIdx.x * 256 + threadIdx.x;
  const int oc = tid & 63;
  const int r  = tid >> 6;            // row 0..ROWS-1
  const int b  = r >> 14;             // r / (NG*GS)
  const int g  = r >> 5;              // global group b*NG+m
  const int n  = knn[r];
  const float x0 = pts[((size_t)b*N_ + n)*3+0] - cent[g*3+0];
  const float x1 = pts[((size_t)b*N_ + n)*3+1] - cent[g*3+1];
  const float x2 = pts[((size_t)b*N_ + n)*3+2] - cent[g*3+2];
  h1[(size_t)r*64 + oc] = W1[oc*3+0]*x0 + W1[oc*3+1]*x1 + W1[oc*3+2]*x2 + b1[oc];
}

// ---------------- deterministic BN stats: per-block partial sums ------------
template<int C>
__global__ __launch_bounds__(256) void stats_kernel(const float* __restrict__ h,
                                                    float* __restrict__ part) {
  constexpr int RPB = 256 / C;
  const int c   = threadIdx.x % C;
  const int sub = threadIdx.x / C;
  float s = 0.f, ss = 0.f;
  for (long r = (long)blockIdx.x * RPB + sub; r < ROWS; r += (long)gridDim.x * RPB) {
    float v = h[r*C + c];
    s += v; ss += v*v;
  }
  __shared__ float shS[256], shQ[256];
  shS[threadIdx.x] = s; shQ[threadIdx.x] = ss;
  __syncthreads();
  if (sub == 0) {
#pragma unroll
    for (int u = 1; u < RPB; ++u) { s += shS[u*C + c]; ss += shQ[u*C + c]; }
    part[(blockIdx.x*2 + 0)*C + c] = s;
    part[(blockIdx.x*2 + 1)*C + c] = ss;
  }
}

template<int C>
__global__ void finalize_kernel(const float* __restrict__ part, int nblk,
    const float* __restrict__ g, const float* __restrict__ be,
    float* __restrict__ scale, float* __restrict__ shift) {
  const int c = threadIdx.x;
  if (c >= C) return;
  float s = 0.f, ss = 0.f;
  for (int i = 0; i < nblk; ++i) { s += part[(i*2+0)*C + c]; ss += part[(i*2+1)*C + c]; }
  const float inv = 1.0f / (float)ROWS;
  float mean = s * inv;
  float var  = ss * inv - mean*mean;
  float sc = g[c] * rsqrtf(var + BN_EPS);
  scale[c] = sc;
  shift[c] = be[c] - mean * sc;
}

// ---------------- GEMM2: (ROWS x 64) @ (64 x 128) via f32 WMMA --------------
__global__ __launch_bounds__(256) void gemm2_kernel(const float* __restrict__ h1,
    const float* __restrict__ scale1, const float* __restrict__ shift1,
    const float* __restrict__ W2, const float* __restrict__ b2,
    float* __restrict__ h2) {
  const int lane = threadIdx.x & 31;
  const int wave = threadIdx.x >> 5;
  const int rowtile = blockIdx.x * 8 + wave;     // 0..8191
  const int coltile = blockIdx.y;                // 0..7
  const int mrow = rowtile*16 + (lane & 15);
  const int ncol = coltile*16 + (lane & 15);
  const int koff = (lane < 16) ? 0 : 2;
  const float* arow = h1 + (size_t)mrow * 64;
  const float* brow = W2 + (size_t)ncol * 64;    // W2[o][c], o-major
  v8f acc = {0.f,0.f,0.f,0.f,0.f,0.f,0.f,0.f};
  for (int kk = 0; kk < 16; ++kk) {
    const int c0 = kk*4 + koff;
    v2f a, bb;
    a.x = fmaxf(0.f, arow[c0]   * scale1[c0]   + shift1[c0]);
    a.y = fmaxf(0.f, arow[c0+1] * scale1[c0+1] + shift1[c0+1]);
    bb.x = brow[c0];
    bb.y = brow[c0+1];
    acc = __builtin_amdgcn_wmma_f32_16x16x4_f32(false, a, false, bb, (short)0, acc, false, false);
  }
  const int rbase = rowtile*16 + ((lane < 16) ? 0 : 8);
  const float bias = b2[ncol];
#pragma unroll
  for (int v = 0; v < 8; ++v)
    h2[(size_t)(rbase + v)*128 + ncol] = acc[v] + bias;
}

// ---- GEMM3: (ROWS x 128) @ (128 x 384), fused bias + max/min + stats -------
__global__ __launch_bounds__(256) void gemm3_kernel(const float* __restrict__ h2,
    const float* __restrict__ scale2, const float* __restrict__ shift2,
    const float* __restrict__ W3, const float* __restrict__ b3,
    float* __restrict__ gmax, float* __restrict__ gmin,
    float* __restrict__ partS, float* __restrict__ partSS) {
  const int lane = threadIdx.x & 31;
  const int wave = threadIdx.x >> 5;
  const int rowtile = blockIdx.x * 8 + wave;     // 0..8191
  const int coltile = blockIdx.y;                // 0..23
  const int mrow = rowtile*16 + (lane & 15);
  const int ncol = coltile*16 + (lane & 15);
  const int koff = (lane < 16) ? 0 : 2;
  const float* arow = h2 + (size_t)mrow * 128;
  const float* brow = W3 + (size_t)ncol * 128;
  v8f acc = {0.f,0.f,0.f,0.f,0.f,0.f,0.f,0.f};
  for (int kk = 0; kk < 32; ++kk) {
    const int c0 = kk*4 + koff;
    v2f a, bb;
    a.x = fmaxf(0.f, arow[c0]   * scale2[c0]   + shift2[c0]);
    a.y = fmaxf(0.f, arow[c0+1] * scale2[c0+1] + shift2[c0+1]);
    bb.x = brow[c0];
    bb.y = brow[c0+1];
    acc = __builtin_amdgcn_wmma_f32_16x16x4_f32(false, a, false, bb, (short)0, acc, false, false);
  }
  const float bias = b3[ncol];
  float vmax = -3.402823466e38f, vmin = 3.402823466e38f, s = 0.f, ss = 0.f;
#pragma unroll
  for (int v = 0; v < 8; ++v) {
    float val = acc[v] + bias;
    vmax = fmaxf(vmax, val); vmin = fminf(vmin, val);
    s += val; ss += val*val;
  }
  // combine the two half-wave row sets (same column)
  vmax = fmaxf(vmax, __shfl_xor(vmax, 16, 32));
  vmin = fminf(vmin, __shfl_xor(vmin, 16, 32));
  s  += __shfl_xor(s, 16, 32);
  ss += __shfl_xor(ss, 16, 32);
  if (lane < 16) {
    const int grp = rowtile >> 1;                // 0..4095 = b*NG+m
    atomicMaxFloat(&gmax[(size_t)grp*EMB + ncol], vmax);
    atomicMinFloat(&gmin[(size_t)grp*EMB + ncol], vmin);
    partS [(size_t)rowtile*EMB + ncol] = s;
    partSS[(size_t)rowtile*EMB + ncol] = ss;
  }
}

// ------------- layer3 stats reduce + finalize (one block per channel) -------
__global__ __launch_bounds__(256) void finalize3_kernel(const float* __restrict__ partS,
    const float* __restrict__ partSS, const float* __restrict__ g,
    const float* __restrict__ be, float* __restrict__ scale, float* __restrict__ shift) {
  const int c = blockIdx.x;
  float s = 0.f, ss = 0.f;
  for (int rt = threadIdx.x; rt < 8192; rt += 256) {
    s  += partS [(size_t)rt*EMB + c];
    ss += partSS[(size_t)rt*EMB + c];
  }
  __shared__ float shS[256], shQ[256];
  shS[threadIdx.x] = s; shQ[threadIdx.x] = ss;
  __syncthreads();
  for (int st = 128; st > 0; st >>= 1) {
    if (threadIdx.x < st) { shS[threadIdx.x] += shS[threadIdx.x+st]; shQ[threadIdx.x] += shQ[threadIdx.x+st]; }
    __syncthreads();
  }
  if (threadIdx.x == 0) {
    const float inv = 1.0f / (float)ROWS;
    float mean = shS[0] * inv;
    float var  = shQ[0] * inv - mean*mean;
    float sc = g[c] * rsqrtf(var + BN_EPS);
    scale[c] = sc;
    shift[c] = be[c] - mean * sc;
  }
}

// ------------- final tokens: BN affine of max (or min if scale<0) -----------
__global__ void tokens_kernel(const float* __restrict__ gmax, const float* __restrict__ gmin,
    const float* __restrict__ scale, const float* __restrict__ shift,
    float* __restrict__ out) {
  int i = blockIdx.x * 256 + threadIdx.x;
  if (i >= B_*NG*EMB) return;
  int c = i % EMB;
  float sc = scale[c];
  float v = (sc >= 0.f) ? gmax[i] : gmin[i];
  out[i] = v * sc + shift[c];
}

// ---------------------------------------------------------------------------
extern "C" void kernel_launch(void* const* d_in, const int* in_sizes, int n_in,
                              void* d_out, int out_size, void* d_ws, size_t ws_size,
                              hipStream_t stream) {
  const float* pts = (const float*)d_in[0];
  const float* W1  = (const float*)d_in[1];
  const float* b1  = (const float*)d_in[2];
  const float* g1  = (const float*)d_in[3];
  const float* be1 = (const float*)d_in[4];
  const float* W2  = (const float*)d_in[5];
  const float* b2  = (const float*)d_in[6];
  const float* g2  = (const float*)d_in[7];
  const float* be2 = (const float*)d_in[8];
  const float* W3  = (const float*)d_in[9];
  const float* b3  = (const float*)d_in[10];
  const float* g3  = (const float*)d_in[11];
  const float* be3 = (const float*)d_in[12];

  float* out_tokens = (float*)d_out;
  float* out_cent   = out_tokens + (size_t)B_*NG*EMB;

  char* w = (char*)d_ws;
  auto alloc = [&](size_t bytes) -> void* {
    void* p = (void*)w; w += (bytes + 255) & ~(size_t)255; return p;
  };
  float* cent   = (float*)alloc((size_t)B_*NG*3*sizeof(float));
  int*   knn    = (int*)  alloc((size_t)ROWS*sizeof(int));
  float* h1     = (float*)alloc((size_t)ROWS*64*sizeof(float));
  float* h2     = (float*)alloc((size_t)ROWS*128*sizeof(float));
  float* gmaxb  = (float*)alloc((size_t)B_*NG*EMB*sizeof(float));
  float* gminb  = (float*)alloc((size_t)B_*NG*EMB*sizeof(float));
  float* partS  = (float*)alloc((size_t)8192*EMB*sizeof(float));
  float* partSS = (float*)alloc((size_t)8192*EMB*sizeof(float));
  float* part1  = (float*)alloc((size_t)512*2*64*sizeof(float));
  float* part2  = (float*)alloc((size_t)512*2*128*sizeof(float));
  float* scale1 = (float*)alloc(64*sizeof(float));
  float* shift1 = (float*)alloc(64*sizeof(float));
  float* scale2 = (float*)alloc(128*sizeof(float));
  float* shift2 = (float*)alloc(128*sizeof(float));
  float* scale3 = (float*)alloc(EMB*sizeof(float));
  float* shift3 = (float*)alloc(EMB*sizeof(float));

  init_kernel<<<(B_*NG*EMB + 255)/256, 256, 0, stream>>>(gmaxb, gminb);
  fps_kernel<<<B_, 1024, 0, stream>>>(pts, cent, out_cent);
  knn_kernel<<<B_*NG, 256, 0, stream>>>(pts, cent, knn);
  layer1_kernel<<<(ROWS*64)/256, 256, 0, stream>>>(pts, cent, knn, W1, b1, h1);
  stats_kernel<64><<<512, 256, 0, stream>>>(h1, part1);
  finalize_kernel<64><<<1, 64, 0, stream>>>(part1, 512, g1, be1, scale1, shift1);
  gemm2_kernel<<<dim3(1024, 8), 256, 0, stream>>>(h1, scale1, shift1, W2, b2, h2);
  stats_kernel<128><<<512, 256, 0, stream>>>(h2, part2);
  finalize_kernel<128><<<1, 128, 0, stream>>>(part2, 512, g2, be2, scale2, shift2);
  gemm3_kernel<<<dim3(1024, 24), 256, 0, stream>>>(h2, scale2, shift2, W3, b3,
                                                   gmaxb, gminb, partS, partSS);
  finalize3_kernel<<<EMB, 256, 0, stream>>>(partS, partSS, g3, be3, scale3, shift3);
  tokens_kernel<<<(B_*NG*EMB + 255)/256, 256, 0, stream>>>(gmaxb, gminb, scale3, shift3, out_tokens);
}